// HMMModel_2216203125035
// MI455X (gfx1250) — compile-verified
//
#include <hip/hip_runtime.h>
#include <cstddef>

// ---------------------------------------------------------------------------
// HMM Baum-Welch E-step for gfx1250 (MI455X), wave32.
// B=512, T=1024, Z=8, S=5 transition matrices, X=2 emission symbols.
//
// Scans use V_WMMA_F32_16X16X4_F32: per wave, 16 batch elements, the
// Z-contraction with per-batch transition matrix K[s_b] is expressed as a
// single 8x40 @ 40x16 matmul via one-hot expansion over the 5 matrices
// (10 chained f32 WMMAs, K=4 each). A-rows 8..15 duplicate rows 0..7 so the
// 16x16 D mirrors into lanes 16..31 -> per-batch state stays lane-local
// (lane handles batch lane&15, replicated across wave halves; no shuffles).
// ---------------------------------------------------------------------------

#define HB 512
#define HT 1024
#define HZ 8
#define HS 5
#define NINF (-10000000000.0f)

typedef float v8f __attribute__((ext_vector_type(8)));
typedef float v2f __attribute__((ext_vector_type(2)));

// ---------------- forward scan: log_alpha + loglik -------------------------
__global__ __launch_bounds__(32) void hmm_fwd(
    const int* __restrict__ s, const int* __restrict__ x,
    const float* __restrict__ mask, const float* __restrict__ pi,
    const float* __restrict__ P, const float* __restrict__ K,
    float* __restrict__ alpha, float* __restrict__ llk_out) {
  const int lane = threadIdx.x;
  const int half = lane >> 4;          // 0: lanes 0-15, 1: lanes 16-31
  const int bloc = lane & 15;
  const int b    = blockIdx.x * 16 + bloc;
  const int jj   = bloc & 7;           // replicated row index (A rows 8..15 = rows 0..7)

  // Preload A operands: Khat[j, 8s+k] = K[s][j][k]; per WMMA i, lane holds
  // cols {4i+2*half, 4i+1+2*half}.
  v2f aK[10];
#pragma unroll
  for (int i = 0; i < 10; ++i) {
    const int c0 = 4 * i + 2 * half, c1 = c0 + 1;
    aK[i].x = K[(c0 >> 3) * 64 + jj * 8 + (c0 & 7)];
    aK[i].y = K[(c1 >> 3) * 64 + jj * 8 + (c1 & 7)];
  }
  float lp0[HZ], lp1[HZ], prev[HZ];
#pragma unroll
  for (int j = 0; j < HZ; ++j) {
    lp0[j]  = __logf(P[2 * j]);
    lp1[j]  = __logf(P[2 * j + 1]);
    prev[j] = __logf(pi[j]);
  }
  const int*   srow = s    + (size_t)b * HT;
  const int*   xrow = x    + (size_t)b * HT;
  const float* mrow = mask + (size_t)b * HT;

  for (int t = 0; t < HT; ++t) {
    const int   sb = srow[t];
    const int   xb = xrow[t];
    const float m  = mrow[t];

    float M = prev[0];
#pragma unroll
    for (int j = 1; j < HZ; ++j) M = fmaxf(M, prev[j]);
    float e[HZ];
#pragma unroll
    for (int j = 0; j < HZ; ++j) e[j] = __expf(prev[j] - M);

    v8f acc = {0.f, 0.f, 0.f, 0.f, 0.f, 0.f, 0.f, 0.f};
#pragma unroll
    for (int i = 0; i < 10; ++i) {
      // B operand: ehat[8s+k, b] = (s==s_b) ? e[k] : 0 ; lane holds rows
      // {4i+2*half, 4i+1+2*half} for column b = lane&15 (lane-local data).
      v2f bb;
      {
        const int rA = 4 * i, rB = 4 * i + 2;
        const float vA = ((rA >> 3) == sb) ? e[rA & 7] : 0.f;
        const float vB = ((rB >> 3) == sb) ? e[rB & 7] : 0.f;
        bb.x = half ? vB : vA;
      }
      {
        const int rA = 4 * i + 1, rB = 4 * i + 3;
        const float vA = ((rA >> 3) == sb) ? e[rA & 7] : 0.f;
        const float vB = ((rB >> 3) == sb) ? e[rB & 7] : 0.f;
        bb.y = half ? vB : vA;
      }
      acc = __builtin_amdgcn_wmma_f32_16x16x4_f32(false, aK[i], false, bb,
                                                  (short)0, acc, false, false);
    }
    const bool on = (m > 0.5f);
    float nw[HZ];
#pragma unroll
    for (int j = 0; j < HZ; ++j) {
      const float lpx = xb ? lp1[j] : lp0[j];
      nw[j]   = on ? (M + __logf(acc[j]) + lpx) : prev[j];
      prev[j] = nw[j];
    }
    if (half == 0) {
      float4* dst = reinterpret_cast<float4*>(alpha + ((size_t)b * HT + t) * HZ);
      dst[0] = make_float4(nw[0], nw[1], nw[2], nw[3]);
      dst[1] = make_float4(nw[4], nw[5], nw[6], nw[7]);
    }
  }
  // loglik = logsumexp(alpha[:, T-1, :])
  float M = prev[0];
#pragma unroll
  for (int j = 1; j < HZ; ++j) M = fmaxf(M, prev[j]);
  float ssum = 0.f;
#pragma unroll
  for (int j = 0; j < HZ; ++j) ssum += __expf(prev[j] - M);
  if (half == 0) llk_out[b] = M + __logf(ssum);
}

// ---------------- backward scan: log_beta (already minus loglik) -----------
__global__ __launch_bounds__(32) void hmm_bwd(
    const int* __restrict__ s, const int* __restrict__ x,
    const float* __restrict__ mask, const float* __restrict__ P,
    const float* __restrict__ K, const float* __restrict__ llk_in,
    float* __restrict__ beta) {
  const int lane = threadIdx.x;
  const int half = lane >> 4;
  const int bloc = lane & 15;
  const int b    = blockIdx.x * 16 + bloc;
  const int jj   = bloc & 7;   // replicated row index = output state k

  // A operands: Khat2[k, 8s+j] = K[s][j][k]
  v2f aK[10];
#pragma unroll
  for (int i = 0; i < 10; ++i) {
    const int c0 = 4 * i + 2 * half, c1 = c0 + 1;
    aK[i].x = K[(c0 >> 3) * 64 + (c0 & 7) * 8 + jj];
    aK[i].y = K[(c1 >> 3) * 64 + (c1 & 7) * 8 + jj];
  }
  float lp0[HZ], lp1[HZ];
#pragma unroll
  for (int j = 0; j < HZ; ++j) {
    lp0[j] = __logf(P[2 * j]);
    lp1[j] = __logf(P[2 * j + 1]);
  }
  const int*   srow = s    + (size_t)b * HT;
  const int*   xrow = x    + (size_t)b * HT;
  const float* mrow = mask + (size_t)b * HT;
  const float  llk  = llk_in[b];

  float bn[HZ];
#pragma unroll
  for (int j = 0; j < HZ; ++j) bn[j] = 0.f;
  int cs = srow[HT - 1];   // carried transition index (mask blends are exact selects)
  int cx = xrow[HT - 1];   // carried emission index

  if (half == 0) {  // log_beta[:, T-1, :] = 0 - loglik
    float4* dst = reinterpret_cast<float4*>(beta + ((size_t)b * HT + (HT - 1)) * HZ);
    dst[0] = make_float4(-llk, -llk, -llk, -llk);
    dst[1] = make_float4(-llk, -llk, -llk, -llk);
  }

  for (int t = HT - 2; t >= 0; --t) {
    const float m1 = mrow[t + 1];
    const float m0 = mrow[t];
    const int   st = srow[t];
    const int   xt = xrow[t];

    float g[HZ];
#pragma unroll
    for (int j = 0; j < HZ; ++j) g[j] = (cx ? lp1[j] : lp0[j]) + bn[j];
    float M = g[0];
#pragma unroll
    for (int j = 1; j < HZ; ++j) M = fmaxf(M, g[j]);
    float e[HZ];
#pragma unroll
    for (int j = 0; j < HZ; ++j) e[j] = __expf(g[j] - M);

    v8f acc = {0.f, 0.f, 0.f, 0.f, 0.f, 0.f, 0.f, 0.f};
#pragma unroll
    for (int i = 0; i < 10; ++i) {
      v2f bb;
      {
        const int rA = 4 * i, rB = 4 * i + 2;
        const float vA = ((rA >> 3) == cs) ? e[rA & 7] : 0.f;
        const float vB = ((rB >> 3) == cs) ? e[rB & 7] : 0.f;
        bb.x = half ? vB : vA;
      }
      {
        const int rA = 4 * i + 1, rB = 4 * i + 3;
        const float vA = ((rA >> 3) == cs) ? e[rA & 7] : 0.f;
        const float vB = ((rB >> 3) == cs) ? e[rB & 7] : 0.f;
        bb.y = half ? vB : vA;
      }
      acc = __builtin_amdgcn_wmma_f32_16x16x4_f32(false, aK[i], false, bb,
                                                  (short)0, acc, false, false);
    }
    const bool on = (m1 > 0.5f);
    float bt[HZ];
#pragma unroll
    for (int k = 0; k < HZ; ++k) bt[k] = on ? (M + __logf(acc[k])) : bn[k];

    if (half == 0) {
      float4* dst = reinterpret_cast<float4*>(beta + ((size_t)b * HT + t) * HZ);
      dst[0] = make_float4(bt[0] - llk, bt[1] - llk, bt[2] - llk, bt[3] - llk);
      dst[1] = make_float4(bt[4] - llk, bt[5] - llk, bt[6] - llk, bt[7] - llk);
    }
    // carry update AFTER computing bt (reference semantics)
    cs = (m0 > 0.5f) ? st : cs;
    cx = (m0 > 0.5f) ? xt : cx;
#pragma unroll
    for (int k = 0; k < HZ; ++k) bn[k] = bt[k];
  }
}

// ---------------- gamma: elementwise over (b,t) ----------------------------
__global__ __launch_bounds__(256) void hmm_gamma(
    const float* __restrict__ mask, const float* __restrict__ alpha,
    const float* __restrict__ beta, float* __restrict__ gamma) {
  const size_t idx = (size_t)blockIdx.x * blockDim.x + threadIdx.x;  // b*T + t
  if (idx >= (size_t)HB * HT) return;
  const float4* la = reinterpret_cast<const float4*>(alpha + idx * HZ);
  const float4* lb = reinterpret_cast<const float4*>(beta + idx * HZ);
  const float4 a0 = la[0], a1 = la[1], b0 = lb[0], b1 = lb[1];
  float g[HZ] = {a0.x + b0.x, a0.y + b0.y, a0.z + b0.z, a0.w + b0.w,
                 a1.x + b1.x, a1.y + b1.y, a1.z + b1.z, a1.w + b1.w};
  float M = g[0];
#pragma unroll
  for (int j = 1; j < HZ; ++j) M = fmaxf(M, g[j]);
  float ssum = 0.f;
#pragma unroll
  for (int j = 0; j < HZ; ++j) ssum += __expf(g[j] - M);
  const float lse = M + __logf(ssum);
  const bool  off = (mask[idx] == 0.0f);
  float4* dst = reinterpret_cast<float4*>(gamma + idx * HZ);
  dst[0] = make_float4(off ? NINF : g[0] - lse, off ? NINF : g[1] - lse,
                       off ? NINF : g[2] - lse, off ? NINF : g[3] - lse);
  dst[1] = make_float4(off ? NINF : g[4] - lse, off ? NINF : g[5] - lse,
                       off ? NINF : g[6] - lse, off ? NINF : g[7] - lse);
}

// ---------------- xi: elementwise over (b,t), t < T-1 ----------------------
__global__ __launch_bounds__(256) void hmm_xi(
    const int* __restrict__ s, const int* __restrict__ x,
    const float* __restrict__ mask, const float* __restrict__ P,
    const float* __restrict__ K, const float* __restrict__ alpha,
    const float* __restrict__ beta, float* __restrict__ xi) {
  __shared__ float lgK[HS * 64];
  __shared__ float lgP[HZ * 2];
  for (int i = threadIdx.x; i < HS * 64; i += blockDim.x) lgK[i] = __logf(K[i]);
  if (threadIdx.x < HZ * 2) lgP[threadIdx.x] = __logf(P[threadIdx.x]);
  __syncthreads();

  const size_t idx = (size_t)blockIdx.x * blockDim.x + threadIdx.x;
  if (idx >= (size_t)HB * (HT - 1)) return;
  const int b = (int)(idx / (HT - 1));
  const int t = (int)(idx % (HT - 1));
  const size_t bt = (size_t)b * HT + t;

  const int   st = s[bt];
  const int   xn = x[bt + 1];
  const float m1 = mask[bt + 1];

  const float* la = alpha + bt * HZ;
  const float* lb = beta + (bt + 1) * HZ;
  float a[HZ], c[HZ];
#pragma unroll
  for (int j = 0; j < HZ; ++j) a[j] = la[j];
#pragma unroll
  for (int k = 0; k < HZ; ++k) c[k] = lb[k] + lgP[k * 2 + xn];

  const float* lk = lgK + st * 64;
  float v[64];
  float M = -3.402823466e+38f;
#pragma unroll
  for (int j = 0; j < HZ; ++j)
#pragma unroll
    for (int k = 0; k < HZ; ++k) {
      const float val = lk[j * HZ + k] + a[j] + c[k];
      v[j * HZ + k] = val;
      M = fmaxf(M, val);
    }
  float ssum = 0.f;
#pragma unroll
  for (int q = 0; q < 64; ++q) ssum += __expf(v[q] - M);
  const float lse = M + __logf(ssum);
  const bool  off = (m1 == 0.0f);

  float4* dst = reinterpret_cast<float4*>(xi + idx * 64);
#pragma unroll
  for (int q = 0; q < 16; ++q) {
    dst[q] = make_float4(off ? NINF : v[4 * q + 0] - lse,
                         off ? NINF : v[4 * q + 1] - lse,
                         off ? NINF : v[4 * q + 2] - lse,
                         off ? NINF : v[4 * q + 3] - lse);
  }
}

// ---------------------------------------------------------------------------
extern "C" void kernel_launch(void* const* d_in, const int* in_sizes, int n_in,
                              void* d_out, int out_size, void* d_ws, size_t ws_size,
                              hipStream_t stream) {
  const int*   s    = (const int*)d_in[0];
  const int*   x    = (const int*)d_in[1];
  const float* mask = (const float*)d_in[2];
  const float* pi   = (const float*)d_in[3];
  const float* P    = (const float*)d_in[4];
  const float* K    = (const float*)d_in[5];

  float* out   = (float*)d_out;
  float* gamma = out;                                        // (B,T,Z)
  float* xi    = gamma + (size_t)HB * HT * HZ;               // (B,T-1,Z,Z)
  float* alpha = xi + (size_t)HB * (HT - 1) * HZ * HZ;       // (B,T,Z)
  float* beta  = alpha + (size_t)HB * HT * HZ;               // (B,T,Z)
  float* llk   = beta + (size_t)HB * HT * HZ;                // (B,)

  hmm_fwd<<<HB / 16, 32, 0, stream>>>(s, x, mask, pi, P, K, alpha, llk);
  hmm_bwd<<<HB / 16, 32, 0, stream>>>(s, x, mask, P, K, llk, beta);

  const int nG = HB * HT;
  hmm_gamma<<<(nG + 255) / 256, 256, 0, stream>>>(mask, alpha, beta, gamma);
  const int nX = HB * (HT - 1);
  hmm_xi<<<(nX + 255) / 256, 256, 0, stream>>>(s, x, mask, P, K, alpha, beta, xi);
}